// PointNetSetAbstraction_6390911337212
// MI455X (gfx1250) — compile-verified
//
#include <hip/hip_runtime.h>

typedef __attribute__((ext_vector_type(2))) float v2f;
typedef __attribute__((ext_vector_type(8))) float v8f;

#define NB    8
#define NPTS  4096
#define NS    1024
#define KNN   32
#define XSTR  68      // 67 input channels padded to 68 (17 k-steps of 4)
#define CNT_BSK (8.0f * 1024.0f * 32.0f)

#if defined(__HIP_DEVICE_COMPILE__) && defined(__gfx1250__) && __has_builtin(__builtin_amdgcn_wmma_f32_16x16x4_f32)
#define USE_WMMA 1
#else
#define USE_WMMA 0
#endif

// ---------------------------------------------------------------------------
// FPS: one block per batch, 512 threads, dist in regs (8 pts/thread), xyz in LDS.
// argmax key = (f32bits << 32) | (4095 - idx)  -> max picks lowest idx on ties.
// ---------------------------------------------------------------------------
__global__ __launch_bounds__(512) void fps_kernel(const float* __restrict__ xyz,
                                                  int* __restrict__ fps_idx,
                                                  float* __restrict__ out_xyz) {
  __shared__ float sx[NPTS], sy[NPTS], sz[NPTS];
  __shared__ unsigned long long wred[16];
  const int b = blockIdx.x, t = threadIdx.x;
  for (int p = t; p < NPTS; p += 512) {
    sx[p] = xyz[(b * NPTS + p) * 3 + 0];
    sy[p] = xyz[(b * NPTS + p) * 3 + 1];
    sz[p] = xyz[(b * NPTS + p) * 3 + 2];
  }
  float dist[8];
#pragma unroll
  for (int i = 0; i < 8; ++i) dist[i] = 1e10f;
  __syncthreads();
  int far = 0;
  for (int s = 0; s < NS; ++s) {
    if (t == 0) {
      fps_idx[b * NS + s] = far;
      out_xyz[(b * NS + s) * 3 + 0] = sx[far];
      out_xyz[(b * NS + s) * 3 + 1] = sy[far];
      out_xyz[(b * NS + s) * 3 + 2] = sz[far];
    }
    const float cx = sx[far], cy = sy[far], cz = sz[far];
    unsigned long long best = 0ull;
#pragma unroll
    for (int i = 0; i < 8; ++i) {
      const int p = t + i * 512;
      const float dx = sx[p] - cx, dy = sy[p] - cy, dz = sz[p] - cz;
      const float d = dx * dx + dy * dy + dz * dz;
      dist[i] = fminf(dist[i], d);
      const unsigned long long key =
          ((unsigned long long)__float_as_uint(dist[i]) << 32) |
          (unsigned int)(NPTS - 1 - p);
      best = (key > best) ? key : best;
    }
#pragma unroll
    for (int o = 16; o >= 1; o >>= 1) {
      const unsigned long long v = __shfl_xor(best, o);
      best = (v > best) ? v : best;
    }
    if ((t & 31) == 0) wred[t >> 5] = best;
    __syncthreads();
    if (t < 16) {
      best = wred[t];
#pragma unroll
      for (int o = 8; o >= 1; o >>= 1) {
        const unsigned long long v = __shfl_xor(best, o);
        best = (v > best) ? v : best;
      }
      if (t == 0) wred[0] = best;
    }
    __syncthreads();
    far = NPTS - 1 - (int)(wred[0] & 0xffffffffu);
    __syncthreads();  // keep wred[] writes of next iter behind this read
  }
}

// ---------------------------------------------------------------------------
// kNN: one wave per centroid, dist[4096] in LDS, 32 rounds of wave-argmin.
// key = (f32bits << 32) | idx  -> min picks smallest dist, lowest idx on ties.
// ---------------------------------------------------------------------------
#define KNN_WAVES 2
__global__ __launch_bounds__(64) void knn_kernel(const float* __restrict__ xyz,
                                                 const float* __restrict__ newxyz,
                                                 int* __restrict__ knn_idx) {
  __shared__ float sd[KNN_WAVES][NPTS];  // 32 KB
  const int wid = threadIdx.x >> 5, lane = threadIdx.x & 31;
  const int cent = blockIdx.x * KNN_WAVES + wid;
  const int b = cent >> 10, s = cent & (NS - 1);
  const float cx = newxyz[(b * NS + s) * 3 + 0];
  const float cy = newxyz[(b * NS + s) * 3 + 1];
  const float cz = newxyz[(b * NS + s) * 3 + 2];
  float* d = sd[wid];
  for (int p = lane; p < NPTS; p += 32) {
    const float dx = xyz[(b * NPTS + p) * 3 + 0] - cx;
    const float dy = xyz[(b * NPTS + p) * 3 + 1] - cy;
    const float dz = xyz[(b * NPTS + p) * 3 + 2] - cz;
    d[p] = dx * dx + dy * dy + dz * dz;
  }
  __syncthreads();
  for (int k = 0; k < KNN; ++k) {
    unsigned long long best = ~0ull;
    for (int p = lane; p < NPTS; p += 32) {
      const unsigned long long key =
          ((unsigned long long)__float_as_uint(d[p]) << 32) | (unsigned int)p;
      best = (key < best) ? key : best;
    }
#pragma unroll
    for (int o = 16; o >= 1; o >>= 1) {
      const unsigned long long v = __shfl_xor(best, o);
      best = (v < best) ? v : best;
    }
    const int idx = (int)(best & 0xffffffffu);
    if (lane == 0) knn_idx[(b * NS + s) * KNN + k] = idx;
    if ((idx & 31) == lane) d[idx] = 3.0e38f;  // remove winner
    __syncthreads();
  }
}

// ---------------------------------------------------------------------------
// 16xCOUT tile matmul via V_WMMA_F32_16X16X4_F32.
// Xt: 16 x XSTR f32 tile in LDS (row-major). Wg: [COUT][LDW] row-major global.
// A frag (per ISA 16x4 f32 layout): lane -> a.x = X[lane&15][4t+2(lane>>4)]
// B frag (4x16):                    lane -> b.x = W[lane&15+16nt][4t+2(lane>>4)]
// D (16x16 f32): vgpr j -> M = j + 8*(lane>>4), N = (lane&15)+16nt
// ---------------------------------------------------------------------------
template <int KEXT, int COUT, int LDW, bool PAD>
__device__ inline void mm_tile(const float* Xt, const float* __restrict__ Wg,
                               v8f* acc, int lane) {
  const int hi = lane >> 4, lo = lane & 15;
  const v8f zero = {0.f, 0.f, 0.f, 0.f, 0.f, 0.f, 0.f, 0.f};
#pragma unroll
  for (int nt = 0; nt < COUT / 16; ++nt) acc[nt] = zero;
#if USE_WMMA
  for (int t = 0; t < KEXT / 4; ++t) {
    const int k = t * 4 + 2 * hi;
    v2f a;
    a.x = Xt[lo * XSTR + k];
    a.y = Xt[lo * XSTR + k + 1];
#pragma unroll
    for (int nt = 0; nt < COUT / 16; ++nt) {
      const int n = lo + 16 * nt;
      v2f bb;
      bb.x = (!PAD || (k < LDW)) ? Wg[n * LDW + k] : 0.f;
      bb.y = (!PAD || (k + 1 < LDW)) ? Wg[n * LDW + k + 1] : 0.f;
      acc[nt] = __builtin_amdgcn_wmma_f32_16x16x4_f32(
          false, a, false, bb, (short)0, acc[nt], false, false);
    }
  }
#else
  for (int nt = 0; nt < COUT / 16; ++nt) {
    const int n = lo + 16 * nt;
    for (int j = 0; j < 8; ++j) {
      const int m = j + 8 * hi;
      float sum = 0.f;
      for (int c = 0; c < LDW; ++c) sum += Xt[m * XSTR + c] * Wg[n * LDW + c];
      acc[nt][j] += sum;
    }
  }
#endif
}

template <int NT>
__device__ inline void accum_stats(const v8f* acc, float* stS, float* stQ) {
#pragma unroll
  for (int nt = 0; nt < NT; ++nt) {
    float sv = 0.f, sq = 0.f;
#pragma unroll
    for (int j = 0; j < 8; ++j) {
      const float v = acc[nt][j];
      sv += v;
      sq += v * v;
    }
    stS[nt] += sv;
    stQ[nt] += sq;
  }
}

template <int NT>
__device__ inline void bn_relu_store(const v8f* acc, const float* __restrict__ sc,
                                     const float* __restrict__ sh, float* Xt,
                                     int lane) {
  const int hi = lane >> 4, lo = lane & 15;
#pragma unroll
  for (int nt = 0; nt < NT; ++nt) {
    const int n = lo + 16 * nt;
    const float scl = sc[n], shf = sh[n];
#pragma unroll
    for (int j = 0; j < 8; ++j) {
      Xt[(j + 8 * hi) * XSTR + n] = fmaxf(acc[nt][j] * scl + shf, 0.f);
    }
  }
}

// ---------------------------------------------------------------------------
// Chain kernel: wave = one centroid (32 neighbor rows = two 16-row M tiles).
// Recomputes L1..DEPTH from gathered inputs (all traffic stays in L2).
// DEPTH<=3,!FINAL: accumulate per-channel sum/sumsq for BN of layer DEPTH.
// FINAL: apply BN3+ReLU, max over K=32, write [B,128,S].
// ---------------------------------------------------------------------------
template <int DEPTH, bool FINAL>
__global__ __launch_bounds__(256) void chain_kernel(
    const float* __restrict__ xyz, const float* __restrict__ feat,
    const int* __restrict__ knn_idx, const float* __restrict__ newxyz,
    const float* __restrict__ W1, const float* __restrict__ W2,
    const float* __restrict__ W3, const float* __restrict__ sc1,
    const float* __restrict__ sh1, const float* __restrict__ sc2,
    const float* __restrict__ sh2, const float* __restrict__ sc3,
    const float* __restrict__ sh3, float* __restrict__ ssum,
    float* __restrict__ ssqs, float* __restrict__ out_feat) {
  __shared__ float shx[8 * 16 * XSTR];  // 8 waves * 16x68 f32 = 34816 B
  const int wid = threadIdx.x >> 5, lane = threadIdx.x & 31;
  const int hi = lane >> 4, lo = lane & 15;
  const int cent = blockIdx.x * 8 + wid;
  const int b = cent >> 10, s = cent & (NS - 1);
  float* Xt = &shx[wid * 16 * XSTR];

  float stS[8], stQ[8], fmx[8];
#pragma unroll
  for (int i = 0; i < 8; ++i) { stS[i] = 0.f; stQ[i] = 0.f; fmx[i] = -1e30f; }

  for (int t2 = 0; t2 < 2; ++t2) {
    // gather: rows = neighbors t2*16..t2*16+15, cols = [dxyz(3) | feat(64) | pad]
    for (int e = lane; e < 16 * XSTR; e += 32) {  // 1088/32 = 34, no divergence
      const int m = e / XSTR, c = e - m * XSTR;
      const int gi = knn_idx[(b * NS + s) * KNN + t2 * 16 + m];
      float v;
      if (c < 3)
        v = xyz[(b * NPTS + gi) * 3 + c] - newxyz[(b * NS + s) * 3 + c];
      else if (c < 67)
        v = feat[(b * 64 + (c - 3)) * NPTS + gi];
      else
        v = 0.f;
      Xt[m * XSTR + c] = v;
    }
    __syncthreads();

    v8f a1[4];
    mm_tile<68, 64, 67, true>(Xt, W1, a1, lane);
    if (DEPTH == 1) {
      accum_stats<4>(a1, stS, stQ);
    } else {
      bn_relu_store<4>(a1, sc1, sh1, Xt, lane);
      __syncthreads();
      v8f a2[4];
      mm_tile<64, 64, 64, false>(Xt, W2, a2, lane);
      if (DEPTH == 2) {
        accum_stats<4>(a2, stS, stQ);
      } else {
        bn_relu_store<4>(a2, sc2, sh2, Xt, lane);
        __syncthreads();
        v8f a3[8];
        mm_tile<64, 128, 64, false>(Xt, W3, a3, lane);
        if (!FINAL) {
          accum_stats<8>(a3, stS, stQ);
        } else {
#pragma unroll
          for (int nt = 0; nt < 8; ++nt) {
            const int n = lo + 16 * nt;
            const float scl = sc3[n], shf = sh3[n];
#pragma unroll
            for (int j = 0; j < 8; ++j) {
              const float v = fmaxf(a3[nt][j] * scl + shf, 0.f);
              fmx[nt] = fmaxf(fmx[nt], v);
            }
          }
        }
      }
    }
    __syncthreads();  // Xt reused by next tile
  }

  if (!FINAL) {
    const int NT = (DEPTH == 3) ? 8 : 4;
#pragma unroll
    for (int nt = 0; nt < 8; ++nt) {
      if (nt >= NT) break;
      const float s1 = stS[nt] + __shfl_xor(stS[nt], 16);
      const float q1 = stQ[nt] + __shfl_xor(stQ[nt], 16);
      if (lane < 16) {
        atomicAdd(&ssum[lo + 16 * nt], s1);
        atomicAdd(&ssqs[lo + 16 * nt], q1);
      }
    }
  } else {
#pragma unroll
    for (int nt = 0; nt < 8; ++nt) {
      const float m = fmaxf(fmx[nt], __shfl_xor(fmx[nt], 16));
      if (lane < 16) out_feat[(b * 128 + (lo + 16 * nt)) * NS + s] = m;
    }
  }
}

__global__ __launch_bounds__(128) void bn_finalize(const float* __restrict__ ssum,
                                                   const float* __restrict__ ssqs,
                                                   const float* __restrict__ gamma,
                                                   const float* __restrict__ beta,
                                                   float* __restrict__ sc,
                                                   float* __restrict__ sh, int C) {
  const int c = threadIdx.x;
  if (c < C) {
    const float inv = 1.0f / CNT_BSK;
    const float mean = ssum[c] * inv;
    const float var = ssqs[c] * inv - mean * mean;
    const float s = gamma[c] * rsqrtf(var + 1e-5f);
    sc[c] = s;
    sh[c] = beta[c] - mean * s;
  }
}

extern "C" void kernel_launch(void* const* d_in, const int* in_sizes, int n_in,
                              void* d_out, int out_size, void* d_ws, size_t ws_size,
                              hipStream_t stream) {
  const float* xyz  = (const float*)d_in[0];
  const float* feat = (const float*)d_in[1];
  const float* W1 = (const float*)d_in[2];
  const float* g1 = (const float*)d_in[3];
  const float* b1 = (const float*)d_in[4];
  const float* W2 = (const float*)d_in[5];
  const float* g2 = (const float*)d_in[6];
  const float* b2 = (const float*)d_in[7];
  const float* W3 = (const float*)d_in[8];
  const float* g3 = (const float*)d_in[9];
  const float* b3 = (const float*)d_in[10];

  float* out_xyz  = (float*)d_out;                  // [8,1024,3]
  float* out_feat = out_xyz + NB * NS * 3;          // [8,128,1024]

  char* ws = (char*)d_ws;
  int* fps_idx = (int*)ws;            ws += NB * NS * sizeof(int);
  int* knn_idx = (int*)ws;            ws += NB * NS * KNN * sizeof(int);
  float* st = (float*)ws;  // [sum1 64|sq1 64|sum2 64|sq2 64|sum3 128|sq3 128|sc/sh...]
  float* sum1 = st + 0,   *sq1 = st + 64;
  float* sum2 = st + 128, *sq2 = st + 192;
  float* sum3 = st + 256, *sq3 = st + 384;
  float* sc1 = st + 512, *sh1 = st + 576;
  float* sc2 = st + 640, *sh2 = st + 704;
  float* sc3 = st + 768, *sh3 = st + 896;

  hipMemsetAsync(st, 0, 512 * sizeof(float), stream);  // zero BN accumulators

  fps_kernel<<<NB, 512, 0, stream>>>(xyz, fps_idx, out_xyz);
  knn_kernel<<<NB * NS / KNN_WAVES, 32 * KNN_WAVES, 0, stream>>>(xyz, out_xyz, knn_idx);

  const int blocks = NB * NS / 8;  // 8 waves (centroids) per 256-thread block
  chain_kernel<1, false><<<blocks, 256, 0, stream>>>(
      xyz, feat, knn_idx, out_xyz, W1, W2, W3, sc1, sh1, sc2, sh2, sc3, sh3,
      sum1, sq1, out_feat);
  bn_finalize<<<1, 128, 0, stream>>>(sum1, sq1, g1, b1, sc1, sh1, 64);

  chain_kernel<2, false><<<blocks, 256, 0, stream>>>(
      xyz, feat, knn_idx, out_xyz, W1, W2, W3, sc1, sh1, sc2, sh2, sc3, sh3,
      sum2, sq2, out_feat);
  bn_finalize<<<1, 128, 0, stream>>>(sum2, sq2, g2, b2, sc2, sh2, 64);

  chain_kernel<3, false><<<blocks, 256, 0, stream>>>(
      xyz, feat, knn_idx, out_xyz, W1, W2, W3, sc1, sh1, sc2, sh2, sc3, sh3,
      sum3, sq3, out_feat);
  bn_finalize<<<1, 128, 0, stream>>>(sum3, sq3, g3, b3, sc3, sh3, 128);

  chain_kernel<3, true><<<blocks, 256, 0, stream>>>(
      xyz, feat, knn_idx, out_xyz, W1, W2, W3, sc1, sh1, sc2, sh2, sc3, sh3,
      nullptr, nullptr, out_feat);
}